// QueryAttention_53326313947675
// MI455X (gfx1250) — compile-verified
//
#include <hip/hip_runtime.h>
#include <math.h>

// QueryAttention for MI455X (gfx1250, wave32)
// x: [128, 2048, 24, 12] fp32  -> viewed as [B, C, P=6, n=48]
// part_feat: [128, 6, 2048] fp32
// out  = [128, 6, 2048] fp32 ; attn = [128, 6, 48] fp32 (concatenated in d_out)

#define C_DIM   2048
#define NPART   6
#define NSEG    48            // patches per part
#define XPSTR   288           // stride between c rows (6*48 floats)
#define THREADS 384           // 12 wave32
#define NWAVES  12

typedef __attribute__((ext_vector_type(2))) float v2f;
typedef __attribute__((ext_vector_type(4))) float v4f;
typedef __attribute__((ext_vector_type(8))) float v8f;

__global__ __launch_bounds__(THREADS)
void qattn_fused_kernel(const float* __restrict__ x,
                        const float* __restrict__ part_feat,
                        float* __restrict__ out,        // [B*6*2048]
                        float* __restrict__ attn_out)   // [B*6*48]
{
    __shared__ float q_s[C_DIM];        // query vector, 8KB
    __shared__ float red[32][NSEG];     // score partials, 6KB
    __shared__ float attn_s[NSEG];

    const int w = blockIdx.x;           // b*6 + p
    const int b = w / NPART;
    const int p = w % NPART;
    const int t = threadIdx.x;

    const float* xbase = x + (size_t)b * (C_DIM * XPSTR) + (size_t)p * NSEG;
    const float* qbase = part_feat + (size_t)w * C_DIM;

    // ---- stage query into LDS ----
    for (int i = t; i < C_DIM; i += THREADS) q_s[i] = qbase[i];
    __syncthreads();

    // ---- pass 1: scores[n] = sum_c q[c] * x[c,n] ----
    // thread -> (n-quad nq in 0..11, c-stripe cb in 0..31)
    const int nq = t % 12;
    const int cb = t / 12;
    v4f acc = {0.f, 0.f, 0.f, 0.f};
    for (int c = cb; c < C_DIM; c += 32) {
        v4f xv = *(const v4f*)(xbase + (size_t)c * XPSTR + nq * 4); // 16B aligned
        float qv = q_s[c];
        acc += qv * xv;
    }
    #pragma unroll
    for (int j = 0; j < 4; ++j) red[cb][nq * 4 + j] = acc[j];

    // deterministic tree reduction over the 32 c-stripes
    for (int s = 16; s >= 1; s >>= 1) {
        __syncthreads();
        if (cb < s) {
            #pragma unroll
            for (int j = 0; j < 4; ++j)
                red[cb][nq * 4 + j] += red[cb + s][nq * 4 + j];
        }
    }
    __syncthreads();

    // ---- softmax over 48 (serial on thread 0; trivial cost) ----
    if (t == 0) {
        const float scale = 0.0220970869120796f;  // 1/sqrt(2048)
        float m = -3.4e38f;
        #pragma unroll 1
        for (int i = 0; i < NSEG; ++i) {
            float v = red[0][i] * scale;
            red[0][i] = v;
            m = fmaxf(m, v);
        }
        float sum = 0.f;
        #pragma unroll 1
        for (int i = 0; i < NSEG; ++i) {
            float e = __expf(red[0][i] - m);
            attn_s[i] = e;
            sum += e;
        }
        float inv = 1.f / sum;
        float* ag = attn_out + (size_t)w * NSEG;
        #pragma unroll 1
        for (int i = 0; i < NSEG; ++i) {
            float a = attn_s[i] * inv;
            attn_s[i] = a;
            ag[i] = a;
        }
    }
    __syncthreads();

    // ---- pass 2: out[c] = sum_n attn[n] * x[c,n] via V_WMMA_F32_16X16X4_F32 ----
    // D(16x16) = A(16x4: x tile, M=c, K=n) x B(4x16: attn broadcast over N) + C
    // 32-bit A 16x4 layout: lane L holds M=(L&15); VGPR0 = K{0|2}, VGPR1 = K{1|3}
    const int wave = t / 32;
    const int lane = t % 32;
    const int mloc = lane & 15;
    const int koff = (lane >> 4) * 2;

    for (int tile = wave; tile < C_DIM / 16; tile += NWAVES) {
        const int c0 = tile * 16;
        const float* xrow = xbase + (size_t)(c0 + mloc) * XPSTR;
        v8f d = {};
        #pragma unroll
        for (int kk = 0; kk < NSEG; kk += 4) {
            v2f a = *(const v2f*)(xrow + kk + koff);       // 8B aligned
            v2f bb;
            bb.x = attn_s[kk + koff];
            bb.y = attn_s[kk + koff + 1];
            d = __builtin_amdgcn_wmma_f32_16x16x4_f32(
                    /*neg_a=*/false, a, /*neg_b=*/false, bb,
                    /*c_mod=*/(short)0, d, /*reuse_a=*/false, /*reuse_b=*/false);
        }
        // D: VGPR r -> M = r + (lane>>4)*8, value identical across N columns
        if (mloc == 0) {
            float* orow = out + (size_t)w * C_DIM + c0 + (lane >> 4) * 8;
            #pragma unroll
            for (int r = 0; r < 8; ++r) orow[r] = d[r];
        }
    }
}

extern "C" void kernel_launch(void* const* d_in, const int* in_sizes, int n_in,
                              void* d_out, int out_size, void* d_ws, size_t ws_size,
                              hipStream_t stream) {
    const float* x  = (const float*)d_in[0];   // [128,2048,24,12]
    const float* pf = (const float*)d_in[1];   // [128,6,2048]
    float* out  = (float*)d_out;                       // first output: [128,6,2048]
    float* attn = out + (size_t)128 * NPART * C_DIM;   // second output: [128,6,48]

    qattn_fused_kernel<<<128 * NPART, THREADS, 0, stream>>>(x, pf, out, attn);
}